// MultiHeadedSparseAttentionModule_26190710571650
// MI455X (gfx1250) — compile-verified
//
#include <hip/hip_runtime.h>
#include <hip/hip_bf16.h>

#define HID 128
#define NHD 8
#define DHD 16
#define LDS_STRIDE 132   // 128 + 4 pad: breaks the 16-way LDS bank conflict

typedef __attribute__((ext_vector_type(2))) float v2f;
typedef __attribute__((ext_vector_type(8))) float v8f;

// ---------------------------------------------------------------------------
// CSR row-pointer build from sorted row[] (deterministic, no atomics)
// ---------------------------------------------------------------------------
__global__ void rowptr_kernel(const int* __restrict__ row, int* __restrict__ rowptr,
                              int E, int N) {
    int e = blockIdx.x * blockDim.x + threadIdx.x;
    if (e >= E) return;
    int r1 = row[e];
    int r0 = (e == 0) ? -1 : row[e - 1];
    for (int n = r0 + 1; n <= r1; ++n) rowptr[n] = e;
    if (e == E - 1) {
        for (int n = r1 + 1; n <= N; ++n) rowptr[n] = E;
    }
}

// ---------------------------------------------------------------------------
// One wave computes a 16x16 f32 output tile of  O = X @ W^T + b
// A (16x4 f32) layout: lanes 0-15 -> M=lane, K = {0,1}; lanes 16-31 -> M=lane-16, K = {2,3}
// B (4x16 f32) layout (mirrored): lane%16 = N, K = {0,1} / {2,3} by lane group
// D (16x16 f32): vgpr r, lane L -> M = r + 8*(L>=16), N = L%16
// ---------------------------------------------------------------------------
__device__ __forceinline__ void wmma_proj_tile(const float* __restrict__ aLDS,
                                               const float* __restrict__ W,
                                               const float* __restrict__ Bb,
                                               float* __restrict__ O,
                                               int rowbase, int colbase, int lane,
                                               int nrows) {
    int n = lane & 15;
#if __has_builtin(__builtin_amdgcn_wmma_f32_16x16x4_f32)
    int koff = (lane >> 4) * 2;
    const float* brow = W + (size_t)(colbase + n) * HID + koff;
    v8f acc = {0.f, 0.f, 0.f, 0.f, 0.f, 0.f, 0.f, 0.f};
#pragma unroll
    for (int kk = 0; kk < HID; kk += 4) {
        v2f a, b;
        a.x = aLDS[kk];     a.y = aLDS[kk + 1];
        b.x = brow[kk];     b.y = brow[kk + 1];
        acc = __builtin_amdgcn_wmma_f32_16x16x4_f32(
            /*neg_a=*/false, a, /*neg_b=*/false, b,
            /*c_mod=*/(short)0, acc, /*reuse_a=*/false, /*reuse_b=*/false);
    }
#else
    // VALU fallback (same per-lane D layout)
    const float* brow = W + (size_t)(colbase + n) * HID;
    const float* abase = aLDS;                       // points at row (lane&15), col (lane>>4)*2
    v8f acc = {0.f, 0.f, 0.f, 0.f, 0.f, 0.f, 0.f, 0.f};
#pragma unroll 1
    for (int kk = 0; kk < HID; ++kk) {
        float w = brow[kk];
#pragma unroll
        for (int r = 0; r < 8; ++r)
            acc[r] += abase[((lane >> 4) * 8 + r - (lane & 15)) * LDS_STRIDE + kk] * w;
    }
#endif
    float bias = Bb[colbase + n];
    int mbase = (lane >> 4) * 8;
#pragma unroll
    for (int r = 0; r < 8; ++r) {
        int rr = rowbase + mbase + r;
        if (rr < nrows)
            O[(size_t)rr * HID + colbase + n] = acc[r] + bias;
    }
}

// Fused projection: one block = 16 rows of X staged in LDS; 8 waves cover the 8
// 16-wide column tiles; each wave produces the tile for up to three weight sets.
__global__ __launch_bounds__(256) void proj3_wmma(
    const float* __restrict__ X,
    const float* __restrict__ W0, const float* __restrict__ B0, float* __restrict__ O0,
    const float* __restrict__ W1, const float* __restrict__ B1, float* __restrict__ O1,
    const float* __restrict__ W2, const float* __restrict__ B2, float* __restrict__ O2,
    int nrows) {
    __shared__ float xt[16 * LDS_STRIDE];
    int tid = threadIdx.x;
    int rowbase = blockIdx.x * 16;

    // cooperative load of the 16x128 tile (512 float4, 2 per thread)
    for (int t = tid; t < 512; t += 256) {
        int r = t >> 5;
        int cvec = t & 31;
        int rsrc = rowbase + r;
        if (rsrc >= nrows) rsrc = nrows - 1;
        float4 val = ((const float4*)(X + (size_t)rsrc * HID))[cvec];
        float* dst = &xt[r * LDS_STRIDE + cvec * 4];
        dst[0] = val.x; dst[1] = val.y; dst[2] = val.z; dst[3] = val.w;
    }
    __syncthreads();

    int wave = tid >> 5;
    int lane = tid & 31;
    int colbase = wave * 16;
    // A-fragment source: row (lane&15), K offset by lane group
    const float* aLDS = &xt[(lane & 15) * LDS_STRIDE + ((lane >> 4) * 2)];

    wmma_proj_tile(aLDS, W0, B0, O0, rowbase, colbase, lane, nrows);
    if (W1) wmma_proj_tile(aLDS, W1, B1, O1, rowbase, colbase, lane, nrows);
    if (W2) wmma_proj_tile(aLDS, W2, B2, O2, rowbase, colbase, lane, nrows);
}

// ---------------------------------------------------------------------------
// SDDMM: one wave32 per edge. Lane L loads float4 of q[row] and k[col]
// (features 4L..4L+3); head of feature f is f&7 -> even lanes carry heads 0-3,
// odd lanes heads 4-7. xor-tree reduce within parity groups, lanes 0/1 store.
// ---------------------------------------------------------------------------
__global__ __launch_bounds__(256) void sddmm_kernel(
    const float* __restrict__ q, const float* __restrict__ k,
    const int* __restrict__ row, const int* __restrict__ col,
    float* __restrict__ scores, int E) {
    int wave = threadIdx.x >> 5;
    int lane = threadIdx.x & 31;
    int e = blockIdx.x * 8 + wave;
    if (e >= E) return;
    int r = row[e], c = col[e];
    float4 q4 = ((const float4*)(q + (size_t)r * HID))[lane];
    float4 k4 = ((const float4*)(k + (size_t)c * HID))[lane];
    float px = q4.x * k4.x, py = q4.y * k4.y, pz = q4.z * k4.z, pw = q4.w * k4.w;
#pragma unroll
    for (int off = 2; off <= 16; off <<= 1) {
        px += __shfl_xor(px, off, 32);
        py += __shfl_xor(py, off, 32);
        pz += __shfl_xor(pz, off, 32);
        pw += __shfl_xor(pw, off, 32);
    }
    if (lane < 2) {
        // 1/sqrt(DH) = 0.25
        float4 s;
        s.x = px * 0.25f; s.y = py * 0.25f; s.z = pz * 0.25f; s.w = pw * 0.25f;
        ((float4*)(scores + (size_t)e * NHD))[lane] = s;
    }
}

// ---------------------------------------------------------------------------
// Fused sparse softmax + SpMM: one wave32 per destination node.
// Pass 1/2: lane = (edge-slot sub = lane>>3, head = lane&7) -> max, then Z.
// Pass 3: lane owns features 4L..4L+3 of the 128-wide output accumulator.
// ---------------------------------------------------------------------------
__global__ __launch_bounds__(256) void attn_spmm_kernel(
    const float* __restrict__ scores, const int* __restrict__ rowptr,
    const int* __restrict__ col, const float* __restrict__ v,
    float* __restrict__ out, int N) {
    int wave = threadIdx.x >> 5;
    int lane = threadIdx.x & 31;
    int i = blockIdx.x * 8 + wave;
    if (i >= N) return;
    int start = rowptr[i], end = rowptr[i + 1];
    int head = lane & 7;
    int sub = lane >> 3;

    float m = -__builtin_inff();
    for (int e = start + sub; e < end; e += 4)
        m = fmaxf(m, scores[(size_t)e * NHD + head]);
    m = fmaxf(m, __shfl_xor(m, 8, 32));
    m = fmaxf(m, __shfl_xor(m, 16, 32));

    float z = 0.f;
    for (int e = start + sub; e < end; e += 4)
        z += __expf(scores[(size_t)e * NHD + head] - m);
    z += __shfl_xor(z, 8, 32);
    z += __shfl_xor(z, 16, 32);
    float invz = (z > 0.f) ? (1.f / z) : 0.f;

    // broadcast the 4 (m, 1/z) pairs this lane's features need (heads hbase..hbase+3)
    int hbase = (lane & 1) * 4;
    float m0 = __shfl(m, hbase + 0, 32), m1 = __shfl(m, hbase + 1, 32);
    float m2 = __shfl(m, hbase + 2, 32), m3 = __shfl(m, hbase + 3, 32);
    float z0 = __shfl(invz, hbase + 0, 32), z1 = __shfl(invz, hbase + 1, 32);
    float z2 = __shfl(invz, hbase + 2, 32), z3 = __shfl(invz, hbase + 3, 32);

    float ax = 0.f, ay = 0.f, az = 0.f, aw = 0.f;
    for (int e = start; e < end; ++e) {
        int c = col[e];
        float4 s4 = ((const float4*)(scores + (size_t)e * NHD))[lane & 1];
        float4 v4 = ((const float4*)(v + (size_t)c * HID))[lane];
        ax += __expf(s4.x - m0) * z0 * v4.x;
        ay += __expf(s4.y - m1) * z1 * v4.y;
        az += __expf(s4.z - m2) * z2 * v4.z;
        aw += __expf(s4.w - m3) * z3 * v4.w;
    }
    float4 acc; acc.x = ax; acc.y = ay; acc.z = az; acc.w = aw;
    ((float4*)(out + (size_t)i * HID))[lane] = acc;
}

// ---------------------------------------------------------------------------
extern "C" void kernel_launch(void* const* d_in, const int* in_sizes, int n_in,
                              void* d_out, int out_size, void* d_ws, size_t ws_size,
                              hipStream_t stream) {
    const float* h  = (const float*)d_in[0];
    const int*   row = (const int*)d_in[1];
    const int*   col = (const int*)d_in[2];
    const float* Wq = (const float*)d_in[3];
    const float* bq = (const float*)d_in[4];
    const float* Wk = (const float*)d_in[5];
    const float* bk = (const float*)d_in[6];
    const float* Wv = (const float*)d_in[7];
    const float* bv = (const float*)d_in[8];
    const float* Wo = (const float*)d_in[9];
    const float* bo = (const float*)d_in[10];

    int N = in_sizes[0] / HID;
    int E = in_sizes[1];

    float* q      = (float*)d_ws;
    float* k      = q + (size_t)N * HID;
    float* v      = k + (size_t)N * HID;
    float* scores = v + (size_t)N * HID;
    float* attn   = scores + (size_t)E * NHD;
    int*   rowptr = (int*)(attn + (size_t)N * HID);
    float* out    = (float*)d_out;

    rowptr_kernel<<<(E + 255) / 256, 256, 0, stream>>>(row, rowptr, E, N);
    proj3_wmma<<<(N + 15) / 16, 256, 0, stream>>>(h, Wq, bq, q, Wk, bk, k, Wv, bv, v, N);
    sddmm_kernel<<<(E + 7) / 8, 256, 0, stream>>>(q, k, row, col, scores, E);
    attn_spmm_kernel<<<(N + 7) / 8, 256, 0, stream>>>(scores, rowptr, col, v, attn, N);
    proj3_wmma<<<(N + 15) / 16, 256, 0, stream>>>(attn, Wo, bo, out,
                                                  nullptr, nullptr, nullptr,
                                                  nullptr, nullptr, nullptr, N);
}